// Net_63496796504131
// MI455X (gfx1250) — compile-verified
//
#include <hip/hip_runtime.h>
#include <math.h>

typedef float v2f __attribute__((ext_vector_type(2)));
typedef float v8f __attribute__((ext_vector_type(8)));

#define PI_F 3.14159265358979323846f
constexpr int N_ATM = 131072;
constexpr int N_BND = 1048576;
constexpr int N_ANG = 2097152;
constexpr int N_GRAPHS = 256;
constexpr int WAVES = 8;          // 256-thread blocks = 8 wave32
constexpr int ROWS_PER_BLOCK = WAVES * 16;

// ---------------- helpers ----------------
__device__ __forceinline__ float fast_rcp(float x) { return __builtin_amdgcn_rcpf(x); }
__device__ __forceinline__ float sigmf(float x) { return fast_rcp(1.f + __expf(-x)); }
__device__ __forceinline__ float siluf(float x) { return x * sigmf(x); }

struct WF { v2f b[4]; };  // 16x16 weight matrix as 4 B-fragments (4x16 each)

__device__ __forceinline__ WF load_wf(const float* __restrict__ W, int col, int half) {
  WF w;
#pragma unroll
  for (int f = 0; f < 4; ++f) {
    int k = 4 * f + 2 * half;
    v2f t; t.x = W[k * 16 + col]; t.y = W[(k + 1) * 16 + col];
    w.b[f] = t;
  }
  return w;
}

__device__ __forceinline__ void load_af(const float* __restrict__ row, int half, v2f a[4]) {
#pragma unroll
  for (int f = 0; f < 4; ++f) {
    int k = 4 * f + 2 * half;
    a[f] = *(const v2f*)(row + k);   // 8B-aligned (row is 64B aligned, k even)
  }
}

__device__ __forceinline__ v8f wmma_step(v2f a, v2f b, v8f c) {
  // D = A(16x4,f32) * B(4x16,f32) + C(16x16,f32)
  return __builtin_amdgcn_wmma_f32_16x16x4_f32(false, a, false, b, (short)0, c, false, false);
}

__device__ __forceinline__ v8f gemm16(const v2f a[4], const WF& w, v8f c) {
#pragma unroll
  for (int f = 0; f < 4; ++f) c = wmma_step(a[f], w.b[f], c);
  return c;
}

// xor-butterfly inside each 16-lane row via v_permlane16_b32 (no LDS traffic).
// sel words: 4 bits per lane; sel0 = lanes 0-7, sel1 = lanes 8-15.
__device__ __forceinline__ float permxor(float x, unsigned s0, unsigned s1) {
  unsigned u = __builtin_bit_cast(unsigned, x);
  unsigned r = __builtin_amdgcn_permlane16(u, u, s0, s1, false, false);
  return __builtin_bit_cast(float, r);
}
__device__ __forceinline__ float rowsum16(float s) {
  s += permxor(s, 0x67452301u, 0xEFCDAB89u);  // lane ^ 1
  s += permxor(s, 0x54761032u, 0xDCFE98BAu);  // lane ^ 2
  s += permxor(s, 0x32107654u, 0xBA98FEDCu);  // lane ^ 4
  s += permxor(s, 0xFEDCBA98u, 0x76543210u);  // lane ^ 8
  return s;
}

// LayerNorm over 16 features: features live across the 16 lanes of a half-wave.
__device__ __forceinline__ void ln16(v8f& z, float g, float b) {
#pragma unroll
  for (int v = 0; v < 8; ++v) {
    float zz = z[v];
    float s1 = rowsum16(zz);
    float s2 = rowsum16(zz * zz);
    float mu = s1 * (1.f / 16.f);
    float var = s2 * (1.f / 16.f) - mu * mu;
    z[v] = (zz - mu) * rsqrtf(var + 1e-5f) * g + b;
  }
}

// ---------------- encoder kernels ----------------
__global__ void k_enc_atom(const int* __restrict__ xatm,
                           const float* __restrict__ W1, const float* __restrict__ b1,
                           const float* __restrict__ W2, const float* __restrict__ b2,
                           const float* __restrict__ lg, const float* __restrict__ lb,
                           float* __restrict__ out) {
  int tid = threadIdx.x, wave = tid >> 5, lane = tid & 31, col = lane & 15, half = lane >> 4;
  long base = ((long)blockIdx.x * WAVES + wave) * 16;
  int sp = xatm[base + col];
  v2f a[4];
#pragma unroll
  for (int f = 0; f < 4; ++f) {
    int k = 4 * f + 2 * half;
    v2f t;
    t.x = siluf(W1[sp * 16 + k]     + b1[k]);
    t.y = siluf(W1[sp * 16 + k + 1] + b1[k + 1]);
    a[f] = t;
  }
  WF w2 = load_wf(W2, col, half);
  v8f acc = {};
  acc = gemm16(a, w2, acc);
  float bc = b2[col];
#pragma unroll
  for (int v = 0; v < 8; ++v) acc[v] += bc;
  ln16(acc, lg[col], lb[col]);
#pragma unroll
  for (int v = 0; v < 8; ++v) out[(base + v + 8 * half) * 16 + col] = acc[v];
}

__global__ void k_enc_bnd(const float* __restrict__ xb,
                          const float* __restrict__ W1, const float* __restrict__ b1,
                          const float* __restrict__ W2, const float* __restrict__ b2,
                          const float* __restrict__ lg, const float* __restrict__ lb,
                          float* __restrict__ out) {
  __shared__ float T[WAVES][16][17];
  int tid = threadIdx.x, wave = tid >> 5, lane = tid & 31, col = lane & 15, half = lane >> 4;
  long base = ((long)blockIdx.x * WAVES + wave) * 16;
  float xv = xb[base + col] + 1e-5f;
  float rxv = fast_rcp(xv);
  float coef = sqrtf(2.f / 5.f) * rxv;
  float t0 = PI_F * xv / 5.f;
  v2f a[4];
#pragma unroll
  for (int f = 0; f < 4; ++f) {
    int k = 4 * f + 2 * half;
    v2f t;
    t.x = coef * __sinf((float)(k + 1) * t0);
    t.y = coef * __sinf((float)(k + 2) * t0);
    a[f] = t;
  }
  WF w1 = load_wf(W1, col, half), w2 = load_wf(W2, col, half);
  v8f h = {};
  h = gemm16(a, w1, h);
  float b1c = b1[col];
#pragma unroll
  for (int v = 0; v < 8; ++v) T[wave][v + 8 * half][col] = siluf(h[v] + b1c);
  __syncthreads();
  v2f a2[4];
#pragma unroll
  for (int f = 0; f < 4; ++f) {
    int k = 4 * f + 2 * half;
    v2f t; t.x = T[wave][col][k]; t.y = T[wave][col][k + 1];
    a2[f] = t;
  }
  v8f o = {};
  o = gemm16(a2, w2, o);
  float b2c = b2[col];
#pragma unroll
  for (int v = 0; v < 8; ++v) o[v] += b2c;
  ln16(o, lg[col], lb[col]);
#pragma unroll
  for (int v = 0; v < 8; ++v) out[(base + v + 8 * half) * 16 + col] = o[v];
}

__global__ void k_enc_ang(const float* __restrict__ xa, const int* __restrict__ mask,
                          const float* __restrict__ eW1, const float* __restrict__ eb1,
                          const float* __restrict__ eW2, const float* __restrict__ eb2,
                          const float* __restrict__ elg, const float* __restrict__ elb,
                          float* __restrict__ out) {
  __shared__ float T[WAVES][16][17];
  int tid = threadIdx.x, wave = tid >> 5, lane = tid & 31, col = lane & 15, half = lane >> 4;
  long base = ((long)blockIdx.x * WAVES + wave) * 16;
  float xv = xa[base + col];
  const float gB = 15.f / PI_F;          // gaussian(0, pi, 16)
  const float gD = 15.f / (2.f * PI_F);  // gaussian(-pi, pi, 16)
  v2f aB[4], aD[4];
#pragma unroll
  for (int f = 0; f < 4; ++f) {
    int k = 4 * f + 2 * half;
    float cB0 = PI_F * (float)k / 15.f, cB1 = PI_F * (float)(k + 1) / 15.f;
    float cD0 = -PI_F + 2.f * PI_F * (float)k / 15.f, cD1 = -PI_F + 2.f * PI_F * (float)(k + 1) / 15.f;
    v2f tb, td;
    float db0 = gB * (xv - cB0), db1 = gB * (xv - cB1);
    float dd0 = gD * (xv - cD0), dd1 = gD * (xv - cD1);
    tb.x = __expf(-db0 * db0); tb.y = __expf(-db1 * db1);
    td.x = __expf(-dd0 * dd0); td.y = __expf(-dd1 * dd1);
    aB[f] = tb; aD[f] = td;
  }
  // weights: branch B = enc index 2, branch D = enc index 3
  WF w1B = load_wf(eW1 + 2 * 256, col, half), w2B = load_wf(eW2 + 2 * 256, col, half);
  WF w1D = load_wf(eW1 + 3 * 256, col, half), w2D = load_wf(eW2 + 3 * 256, col, half);

  // ---- branch B ----
  v8f h = {};
  h = gemm16(aB, w1B, h);
  float bc = eb1[2 * 16 + col];
#pragma unroll
  for (int v = 0; v < 8; ++v) T[wave][v + 8 * half][col] = siluf(h[v] + bc);
  __syncthreads();
  v2f a2[4];
#pragma unroll
  for (int f = 0; f < 4; ++f) {
    int k = 4 * f + 2 * half;
    v2f t; t.x = T[wave][col][k]; t.y = T[wave][col][k + 1];
    a2[f] = t;
  }
  __syncthreads();
  v8f oB = {};
  oB = gemm16(a2, w2B, oB);
  bc = eb2[2 * 16 + col];
#pragma unroll
  for (int v = 0; v < 8; ++v) oB[v] += bc;
  ln16(oB, elg[2 * 16 + col], elb[2 * 16 + col]);

  // ---- branch D ----
  h = (v8f){};
  h = gemm16(aD, w1D, h);
  bc = eb1[3 * 16 + col];
#pragma unroll
  for (int v = 0; v < 8; ++v) T[wave][v + 8 * half][col] = siluf(h[v] + bc);
  __syncthreads();
#pragma unroll
  for (int f = 0; f < 4; ++f) {
    int k = 4 * f + 2 * half;
    v2f t; t.x = T[wave][col][k]; t.y = T[wave][col][k + 1];
    a2[f] = t;
  }
  v8f oD = {};
  oD = gemm16(a2, w2D, oD);
  bc = eb2[3 * 16 + col];
#pragma unroll
  for (int v = 0; v < 8; ++v) oD[v] += bc;
  ln16(oD, elg[3 * 16 + col], elb[3 * 16 + col]);

#pragma unroll
  for (int v = 0; v < 8; ++v) {
    long r = base + v + 8 * half;
    int md = mask[r];
    out[r * 16 + col] = md ? oD[v] : oB[v];
  }
}

// ---------------- edge-gated conv ----------------
__global__ void k_edge(const float* __restrict__ x, float* __restrict__ e,
                       const int* __restrict__ src, const int* __restrict__ dst,
                       const float* __restrict__ W,      // [5][16][16]
                       const float* __restrict__ b0,
                       const float* __restrict__ lg, const float* __restrict__ lb,
                       float* __restrict__ num, float* __restrict__ den) {
  int tid = threadIdx.x, wave = tid >> 5, lane = tid & 31, col = lane & 15, half = lane >> 4;
  long base = ((long)blockIdx.x * WAVES + wave) * 16;
  int s0 = src[base + col];
  int d0 = dst[base + col];
  v2f axs[4], axd[4], ae[4];
  load_af(x + (long)s0 * 16, half, axs);
  load_af(x + (long)d0 * 16, half, axd);
  load_af(e + (base + col) * 16, half, ae);
  WF w0 = load_wf(W, col, half), w1 = load_wf(W + 256, col, half),
     w2 = load_wf(W + 512, col, half), w4 = load_wf(W + 1024, col, half);
  v8f z = {};
  z = gemm16(axs, w0, z);
  z = gemm16(axd, w1, z);
  z = gemm16(ae,  w2, z);
  v8f mm = {};
  mm = gemm16(axs, w4, mm);
  float b0c = b0[col];
  v8f sig, msg;
#pragma unroll
  for (int v = 0; v < 8; ++v) {
    float zz = z[v] + b0c;
    z[v] = zz;
    sig[v] = sigmf(zz);
    msg[v] = sig[v] * mm[v];
  }
#pragma unroll
  for (int v = 0; v < 8; ++v) {
    long ev = base + v + 8 * half;
    long dd = dst[ev];
    atomicAdd(&num[dd * 16 + col], msg[v]);
    atomicAdd(&den[dd * 16 + col], sig[v]);
  }
  ln16(z, lg[col], lb[col]);
#pragma unroll
  for (int v = 0; v < 8; ++v) {
    long ev = base + v + 8 * half;
    float eo = e[ev * 16 + col];
    e[ev * 16 + col] = eo + siluf(z[v]);
  }
}

__global__ void k_node(float* __restrict__ x,
                       const float* __restrict__ num, const float* __restrict__ den,
                       const float* __restrict__ W3, const float* __restrict__ b1,
                       const float* __restrict__ lg, const float* __restrict__ lb) {
  int tid = threadIdx.x, wave = tid >> 5, lane = tid & 31, col = lane & 15, half = lane >> 4;
  long base = ((long)blockIdx.x * WAVES + wave) * 16;
  v2f ax[4];
  load_af(x + (base + col) * 16, half, ax);
  WF w3 = load_wf(W3, col, half);
  v8f acc = {};
  acc = gemm16(ax, w3, acc);
  float b1c = b1[col];
#pragma unroll
  for (int v = 0; v < 8; ++v) {
    long r = base + v + 8 * half;
    acc[v] += b1c + num[r * 16 + col] * fast_rcp(den[r * 16 + col] + 1e-5f);
  }
  ln16(acc, lg[col], lb[col]);
#pragma unroll
  for (int v = 0; v < 8; ++v) {
    long r = base + v + 8 * half;
    x[r * 16 + col] += siluf(acc[v]);
  }
}

// ---------------- misc kernels ----------------
__global__ void k_fill(float* __restrict__ p, long n) {
  long i = (long)blockIdx.x * blockDim.x + threadIdx.x;
  long stride = (long)gridDim.x * blockDim.x;
  for (; i < n; i += stride) p[i] = 0.f;
}

__global__ void k_pool(const float* __restrict__ h, const int* __restrict__ batch,
                       float* __restrict__ pooled) {
  long i = (long)blockIdx.x * blockDim.x + threadIdx.x;  // N_ATM*16 threads exactly
  int atom = (int)(i >> 4), c = (int)(i & 15);
  int g = batch[atom];
  atomicAdd(&pooled[g * 16 + c], h[i]);
}

__global__ void k_head(const float* __restrict__ pooled, const float* __restrict__ fp,
                       const float* __restrict__ l1W, const float* __restrict__ l1b,
                       const float* __restrict__ l2W, const float* __restrict__ l2b,
                       float* __restrict__ out) {
  int g = threadIdx.x;  // 256 graphs, one block
  float xv[18];
#pragma unroll
  for (int i = 0; i < 16; ++i) xv[i] = pooled[g * 16 + i];
  xv[16] = fp[2 * g]; xv[17] = fp[2 * g + 1];
  float o = l2b[0];
#pragma unroll
  for (int j = 0; j < 16; ++j) {
    float a = l1b[j];
#pragma unroll
    for (int i = 0; i < 18; ++i) a += xv[i] * l1W[i * 16 + j];
    a = a > 0.f ? a : 0.01f * a;
    o += a * l2W[j];
  }
  out[g] = o;
}

// ---------------- launch ----------------
extern "C" void kernel_launch(void* const* d_in, const int* in_sizes, int n_in,
                              void* d_out, int out_size, void* d_ws, size_t ws_size,
                              hipStream_t stream) {
  const int*   x_atm     = (const int*)d_in[0];
  const float* x_bnd     = (const float*)d_in[1];
  const float* x_ang     = (const float*)d_in[2];
  const int*   mask      = (const int*)d_in[3];
  const int*   eiG       = (const int*)d_in[4];
  const int*   eiA       = (const int*)d_in[5];
  const int*   batch     = (const int*)d_in[6];
  const float* forcepair = (const float*)d_in[7];
  const float* enc_W1    = (const float*)d_in[8];
  const float* enc_b1    = (const float*)d_in[9];
  const float* enc_W2    = (const float*)d_in[10];
  const float* enc_b2    = (const float*)d_in[11];
  const float* enc_ln_g  = (const float*)d_in[12];
  const float* enc_ln_b  = (const float*)d_in[13];
  const float* conv_W    = (const float*)d_in[14];
  const float* conv_b    = (const float*)d_in[15];
  const float* conv_ln   = (const float*)d_in[16];
  const float* l1_W      = (const float*)d_in[17];
  const float* l1_b      = (const float*)d_in[18];
  const float* l2_W      = (const float*)d_in[19];
  const float* l2_b      = (const float*)d_in[20];
  float* out = (float*)d_out;

  const int* srcA = eiA,  * dstA = eiA + N_ANG;
  const int* srcG = eiG,  * dstG = eiG + N_BND;

  // workspace carve
  char* w = (char*)d_ws;
  float* h_atm  = (float*)w;  w += (size_t)N_ATM * 16 * 4;
  float* h_bnd  = (float*)w;  w += (size_t)N_BND * 16 * 4;
  float* h_ang  = (float*)w;  w += (size_t)N_ANG * 16 * 4;
  float* num    = (float*)w;  w += (size_t)N_BND * 16 * 4;
  float* den    = (float*)w;  w += (size_t)N_BND * 16 * 4;
  float* pooled = (float*)w;  w += (size_t)N_GRAPHS * 16 * 4;

  // ---- encoder ----
  k_enc_atom<<<N_ATM / ROWS_PER_BLOCK, 256, 0, stream>>>(
      x_atm, enc_W1 + 0, enc_b1 + 0, enc_W2 + 0, enc_b2 + 0,
      enc_ln_g + 0, enc_ln_b + 0, h_atm);
  k_enc_bnd<<<N_BND / ROWS_PER_BLOCK, 256, 0, stream>>>(
      x_bnd, enc_W1 + 256, enc_b1 + 16, enc_W2 + 256, enc_b2 + 16,
      enc_ln_g + 16, enc_ln_b + 16, h_bnd);
  k_enc_ang<<<N_ANG / ROWS_PER_BLOCK, 256, 0, stream>>>(
      x_ang, mask, enc_W1, enc_b1, enc_W2, enc_b2, enc_ln_g, enc_ln_b, h_ang);

  // ---- 3 conv rounds ----
  for (int c = 0; c < 3; ++c) {
    const float* Wc0  = conv_W  + (size_t)(c * 2 + 0) * 5 * 256;
    const float* bc0  = conv_b  + (size_t)(c * 2 + 0) * 2 * 16;
    const float* lnc0 = conv_ln + (size_t)(c * 2 + 0) * 4 * 16;  // [which][g/b][16]
    const float* Wc1  = conv_W  + (size_t)(c * 2 + 1) * 5 * 256;
    const float* bc1  = conv_b  + (size_t)(c * 2 + 1) * 2 * 16;
    const float* lnc1 = conv_ln + (size_t)(c * 2 + 1) * 4 * 16;

    // line graph: nodes = bonds, edges = angles
    k_fill<<<8192, 256, 0, stream>>>(num, 2L * N_BND * 16);  // num & den contiguous
    k_edge<<<N_ANG / ROWS_PER_BLOCK, 256, 0, stream>>>(
        h_bnd, h_ang, srcA, dstA, Wc0, bc0,
        lnc0 + 32, lnc0 + 48, num, den);
    k_node<<<N_BND / ROWS_PER_BLOCK, 256, 0, stream>>>(
        h_bnd, num, den, Wc0 + 768, bc0 + 16, lnc0 + 0, lnc0 + 16);

    // atom graph: nodes = atoms, edges = bonds
    float* numA = num;
    float* denA = num + (size_t)N_ATM * 16;
    k_fill<<<2048, 256, 0, stream>>>(numA, 2L * N_ATM * 16);
    k_edge<<<N_BND / ROWS_PER_BLOCK, 256, 0, stream>>>(
        h_atm, h_bnd, srcG, dstG, Wc1, bc1,
        lnc1 + 32, lnc1 + 48, numA, denA);
    k_node<<<N_ATM / ROWS_PER_BLOCK, 256, 0, stream>>>(
        h_atm, numA, denA, Wc1 + 768, bc1 + 16, lnc1 + 0, lnc1 + 16);
  }

  // ---- pool + head ----
  k_fill<<<16, 256, 0, stream>>>(pooled, (long)N_GRAPHS * 16);
  k_pool<<<(N_ATM * 16) / 256, 256, 0, stream>>>(h_atm, batch, pooled);
  k_head<<<1, 256, 0, stream>>>(pooled, forcepair, l1_W, l1_b, l2_W, l2_b, out);
}